// FujiSparseMoE_2611340116637
// MI455X (gfx1250) — compile-verified
//
#include <hip/hip_runtime.h>
#include <math.h>

// ---------------------------------------------------------------------------
// FujiSparseMoE for MI455X (gfx1250).  fp32 reference -> V_WMMA_F32_16X16X4_F32
// Register-blocked: each wave owns a 32x32 (fused g/u) or 32x64 macro-tile so
// A/B fragments are reused across tiles (6 b64 loads per 8 WMMA).
// ---------------------------------------------------------------------------

typedef __attribute__((ext_vector_type(2))) float v2f;
typedef __attribute__((ext_vector_type(8))) float v8f;

#define T_TOK 4096      // S*B
#define DIM   1024      // D
#define NEXP  64        // E
#define TOPK  2         // K
#define IDIM  512       // I
#define SIDIM 512       // SI
#define CAP   256       // per-expert capacity

__device__ __forceinline__ v8f wmma4(v2f a, v2f b, v8f c) {
  // D = A(16x4 f32) * B(4x16 f32) + C(16x16 f32)
  return __builtin_amdgcn_wmma_f32_16x16x4_f32(false, a, false, b, (short)0, c,
                                               false, false);
}

__device__ __forceinline__ float silu_mul(float g, float u) {
  return (g / (1.0f + expf(-g))) * u;
}

// ---------------------------------------------------------------------------
// 1) Router: logits -> softmax (fp32) -> top-2 -> normalized weights.
// ---------------------------------------------------------------------------
__global__ __launch_bounds__(64) void router_kernel(
    const float* __restrict__ x, const float* __restrict__ rw,
    float* __restrict__ probs_out, int* __restrict__ top_i,
    float* __restrict__ top_w) {
  __shared__ float xr[DIM];
  __shared__ float lg[NEXP];
  const int t = blockIdx.x;
  const int tid = threadIdx.x;

  for (int i = tid; i < DIM; i += 64) xr[i] = x[(size_t)t * DIM + i];
  __syncthreads();

  const float* wrow = rw + (size_t)tid * DIM;
  float s = 0.f;
  for (int k = 0; k < DIM; ++k) s += xr[k] * wrow[k];
  lg[tid] = s;
  __syncthreads();

  if (tid == 0) {
    float mx = lg[0];
    for (int e = 1; e < NEXP; ++e) mx = fmaxf(mx, lg[e]);
    float sum = 0.f;
    for (int e = 0; e < NEXP; ++e) { float p = expf(lg[e] - mx); lg[e] = p; sum += p; }
    float inv = 1.0f / sum;
    for (int e = 0; e < NEXP; ++e) lg[e] *= inv;       // lg[] now holds probs
    // top-2 (ties -> lower index first, matches lax.top_k)
    int i0 = 0; float v0 = lg[0];
    for (int e = 1; e < NEXP; ++e) if (lg[e] > v0) { v0 = lg[e]; i0 = e; }
    int i1 = (i0 == 0) ? 1 : 0; float v1 = lg[i1];
    for (int e = 0; e < NEXP; ++e)
      if (e != i0 && lg[e] > v1) { v1 = lg[e]; i1 = e; }
    float wsum = v0 + v1 + 1e-9f;
    top_i[t * TOPK + 0] = i0;  top_w[t * TOPK + 0] = v0 / wsum;
    top_i[t * TOPK + 1] = i1;  top_w[t * TOPK + 1] = v1 / wsum;
  }
  __syncthreads();
  probs_out[(size_t)t * NEXP + tid] = lg[tid];
}

// ---------------------------------------------------------------------------
// 2) Dispatch: thread e scans pairs in order -> capacity-clamped slot lists.
// ---------------------------------------------------------------------------
__global__ __launch_bounds__(64) void dispatch_kernel(
    const int* __restrict__ top_i, const float* __restrict__ top_w,
    int* __restrict__ counts, int* __restrict__ tok_list,
    int* __restrict__ pair_list, float* __restrict__ w_list) {
  const int e = threadIdx.x;
  int cnt = 0;
  for (int p = 0; p < T_TOK * TOPK; ++p) {
    if (top_i[p] == e && cnt < CAP) {
      tok_list[e * CAP + cnt]  = p >> 1;   // source token
      pair_list[e * CAP + cnt] = p;        // flat (token, k) pair index
      w_list[e * CAP + cnt]    = top_w[p];
      ++cnt;
    }
  }
  counts[e] = cnt;
}

__global__ void zero_kernel(float* __restrict__ p, int n) {
  int i = blockIdx.x * blockDim.x + threadIdx.x;
  if (i < n) p[i] = 0.0f;
}

// ---------------------------------------------------------------------------
// Fused SwiGLU up-projection macro-kernel body:
//   wave computes rows [m] x cols [n0..n0+31] of silu(A@Wg^T)*(A@Wu^T).
//   2 M-tiles x 2 N-tiles x {g,u} = 8 accumulators; 6 loads per 8 WMMA.
//   A rows given by two row pointers (direct or token-gathered).
// ---------------------------------------------------------------------------
__device__ __forceinline__ void swiglu_tile(
    const float* arow0, const float* arow1,          // A rows (lane-resolved)
    const float* g0, const float* g1,                // Wg rows n0+lr, n0+16+lr
    const float* u0, const float* u1,                // Wu rows n0+lr, n0+16+lr
    int kdim, int sub, v8f accg[2][2], v8f accu[2][2]) {
  for (int k0 = 0; k0 < kdim; k0 += 16) {
    __builtin_prefetch(g0 + k0 + 256, 0, 1);
    __builtin_prefetch(g1 + k0 + 256, 0, 1);
    __builtin_prefetch(u0 + k0 + 256, 0, 1);
    __builtin_prefetch(u1 + k0 + 256, 0, 1);
#pragma unroll
    for (int kk = 0; kk < 16; kk += 4) {
      const int ko = k0 + kk + 2 * sub;
      v2f a0  = *(const v2f*)(arow0 + ko);
      v2f a1  = *(const v2f*)(arow1 + ko);
      v2f bg0 = *(const v2f*)(g0 + ko);
      v2f bg1 = *(const v2f*)(g1 + ko);
      v2f bu0 = *(const v2f*)(u0 + ko);
      v2f bu1 = *(const v2f*)(u1 + ko);
      accg[0][0] = wmma4(a0, bg0, accg[0][0]);
      accg[0][1] = wmma4(a0, bg1, accg[0][1]);
      accg[1][0] = wmma4(a1, bg0, accg[1][0]);
      accg[1][1] = wmma4(a1, bg1, accg[1][1]);
      accu[0][0] = wmma4(a0, bu0, accu[0][0]);
      accu[0][1] = wmma4(a0, bu1, accu[0][1]);
      accu[1][0] = wmma4(a1, bu0, accu[1][0]);
      accu[1][1] = wmma4(a1, bu1, accu[1][1]);
    }
  }
}

// ---------------------------------------------------------------------------
// Plain GEMM macro-kernel body: rows [m] x cols [n0..n0+63] of A@W^T.
//   2 M-tiles x 4 N-tiles = 8 accumulators; 6 loads per 8 WMMA.
// ---------------------------------------------------------------------------
__device__ __forceinline__ void gemm_tile(
    const float* arow0, const float* arow1,
    const float* b0, const float* b1, const float* b2, const float* b3,
    int kdim, int sub, v8f acc[2][4]) {
  for (int k0 = 0; k0 < kdim; k0 += 16) {
    __builtin_prefetch(b0 + k0 + 256, 0, 1);
    __builtin_prefetch(b1 + k0 + 256, 0, 1);
    __builtin_prefetch(b2 + k0 + 256, 0, 1);
    __builtin_prefetch(b3 + k0 + 256, 0, 1);
#pragma unroll
    for (int kk = 0; kk < 16; kk += 4) {
      const int ko = k0 + kk + 2 * sub;
      v2f a0 = *(const v2f*)(arow0 + ko);
      v2f a1 = *(const v2f*)(arow1 + ko);
      v2f f0 = *(const v2f*)(b0 + ko);
      v2f f1 = *(const v2f*)(b1 + ko);
      v2f f2 = *(const v2f*)(b2 + ko);
      v2f f3 = *(const v2f*)(b3 + ko);
      acc[0][0] = wmma4(a0, f0, acc[0][0]);
      acc[0][1] = wmma4(a0, f1, acc[0][1]);
      acc[0][2] = wmma4(a0, f2, acc[0][2]);
      acc[0][3] = wmma4(a0, f3, acc[0][3]);
      acc[1][0] = wmma4(a1, f0, acc[1][0]);
      acc[1][1] = wmma4(a1, f1, acc[1][1]);
      acc[1][2] = wmma4(a1, f2, acc[1][2]);
      acc[1][3] = wmma4(a1, f3, acc[1][3]);
    }
  }
}

// ---------------------------------------------------------------------------
// 3) Shared expert, stage 1: s1 = silu(x@Wg^T) * (x@Wu^T).
//    Wave macro-tile: 32 rows x 32 cols.
// ---------------------------------------------------------------------------
__global__ __launch_bounds__(32) void shared_mlp1_kernel(
    const float* __restrict__ x, const float* __restrict__ Wg,
    const float* __restrict__ Wu, float* __restrict__ s1) {
  const int lane = threadIdx.x & 31;
  const int lr = lane & 15, sub = lane >> 4;
  const int n0 = blockIdx.x * 32;
  const int m0 = blockIdx.y * 32;
  const float* arow0 = x + (size_t)(m0 + lr) * DIM;
  const float* arow1 = x + (size_t)(m0 + 16 + lr) * DIM;
  const float* g0 = Wg + (size_t)(n0 + lr) * DIM;
  const float* g1 = Wg + (size_t)(n0 + 16 + lr) * DIM;
  const float* u0 = Wu + (size_t)(n0 + lr) * DIM;
  const float* u1 = Wu + (size_t)(n0 + 16 + lr) * DIM;
  v8f accg[2][2] = {}; v8f accu[2][2] = {};
  swiglu_tile(arow0, arow1, g0, g1, u0, u1, DIM, sub, accg, accu);
#pragma unroll
  for (int mi = 0; mi < 2; ++mi)
#pragma unroll
    for (int ni = 0; ni < 2; ++ni)
#pragma unroll
      for (int v = 0; v < 8; ++v) {
        int row = m0 + 16 * mi + v + 8 * sub;
        int col = n0 + 16 * ni + lr;
        s1[(size_t)row * SIDIM + col] = silu_mul(accg[mi][ni][v], accu[mi][ni][v]);
      }
}

// ---------------------------------------------------------------------------
// 4) Shared expert, stage 2: out_main = s1 @ Wd^T  (written into d_out).
//    Wave macro-tile: 32 rows x 64 cols.
// ---------------------------------------------------------------------------
__global__ __launch_bounds__(32) void shared_mlp2_kernel(
    const float* __restrict__ s1, const float* __restrict__ Wd,
    float* __restrict__ out) {
  const int lane = threadIdx.x & 31;
  const int lr = lane & 15, sub = lane >> 4;
  const int n0 = blockIdx.x * 64;
  const int m0 = blockIdx.y * 32;
  const float* arow0 = s1 + (size_t)(m0 + lr) * SIDIM;
  const float* arow1 = s1 + (size_t)(m0 + 16 + lr) * SIDIM;
  const float* b0 = Wd + (size_t)(n0 + lr) * SIDIM;
  const float* b1 = Wd + (size_t)(n0 + 16 + lr) * SIDIM;
  const float* b2 = Wd + (size_t)(n0 + 32 + lr) * SIDIM;
  const float* b3 = Wd + (size_t)(n0 + 48 + lr) * SIDIM;
  v8f acc[2][4] = {};
  gemm_tile(arow0, arow1, b0, b1, b2, b3, SIDIM, sub, acc);
#pragma unroll
  for (int mi = 0; mi < 2; ++mi)
#pragma unroll
    for (int ni = 0; ni < 4; ++ni)
#pragma unroll
      for (int v = 0; v < 8; ++v) {
        int row = m0 + 16 * mi + v + 8 * sub;
        int col = n0 + 16 * ni + lr;
        out[(size_t)row * DIM + col] = acc[mi][ni][v];
      }
}

// ---------------------------------------------------------------------------
// 5) Expert GEMM1: act[e] = silu(buf@Wg^T)*(buf@Wu^T); A rows gathered from x
//    via the token list.  Wave macro-tile: 32 slots x 32 cols.
// ---------------------------------------------------------------------------
__global__ __launch_bounds__(32) void expert_mlp1_kernel(
    const float* __restrict__ x, const float* __restrict__ gate_up,
    const int* __restrict__ counts, const int* __restrict__ tok_list,
    float* __restrict__ act) {
  const int e = blockIdx.z;
  const int count = counts[e];
  const int m0 = blockIdx.y * 32;
  if (m0 >= count) return;
  const int lane = threadIdx.x & 31;
  const int lr = lane & 15, sub = lane >> 4;
  const int n0 = blockIdx.x * 32;

  const int s0i = m0 + lr, s1i = m0 + 16 + lr;
  const int tok0 = (s0i < count) ? tok_list[e * CAP + s0i] : 0;
  const int tok1 = (s1i < count) ? tok_list[e * CAP + s1i] : 0;
  const float* arow0 = x + (size_t)tok0 * DIM;
  const float* arow1 = x + (size_t)tok1 * DIM;
  const float* gup = gate_up + (size_t)e * (2 * IDIM) * DIM;
  const float* g0 = gup + (size_t)(n0 + lr) * DIM;
  const float* g1 = gup + (size_t)(n0 + 16 + lr) * DIM;
  const float* u0 = gup + (size_t)(IDIM + n0 + lr) * DIM;
  const float* u1 = gup + (size_t)(IDIM + n0 + 16 + lr) * DIM;

  v8f accg[2][2] = {}; v8f accu[2][2] = {};
  swiglu_tile(arow0, arow1, g0, g1, u0, u1, DIM, sub, accg, accu);
#pragma unroll
  for (int mi = 0; mi < 2; ++mi)
#pragma unroll
    for (int ni = 0; ni < 2; ++ni)
#pragma unroll
      for (int v = 0; v < 8; ++v) {
        int row = m0 + 16 * mi + v + 8 * sub;   // < CAP always
        int col = n0 + 16 * ni + lr;
        act[((size_t)e * CAP + row) * IDIM + col] =
            silu_mul(accg[mi][ni][v], accu[mi][ni][v]);
      }
}

// ---------------------------------------------------------------------------
// 6) Expert GEMM2: y = act @ Wd^T, weighted scatter to per-pair buffer.
//    Wave macro-tile: 32 slots x 64 cols.  Deterministic, no atomics.
// ---------------------------------------------------------------------------
__global__ __launch_bounds__(32) void expert_mlp2_kernel(
    const float* __restrict__ act, const float* __restrict__ down,
    const int* __restrict__ counts, const int* __restrict__ pair_list,
    const float* __restrict__ w_list, float* __restrict__ y_pair) {
  const int e = blockIdx.z;
  const int count = counts[e];
  const int m0 = blockIdx.y * 32;
  if (m0 >= count) return;
  const int lane = threadIdx.x & 31;
  const int lr = lane & 15, sub = lane >> 4;
  const int n0 = blockIdx.x * 64;

  const float* abase = act + (size_t)e * CAP * IDIM;
  const float* arow0 = abase + (size_t)(m0 + lr) * IDIM;
  const float* arow1 = abase + (size_t)(m0 + 16 + lr) * IDIM;
  const float* wbase = down + (size_t)e * DIM * IDIM;
  const float* b0 = wbase + (size_t)(n0 + lr) * IDIM;
  const float* b1 = wbase + (size_t)(n0 + 16 + lr) * IDIM;
  const float* b2 = wbase + (size_t)(n0 + 32 + lr) * IDIM;
  const float* b3 = wbase + (size_t)(n0 + 48 + lr) * IDIM;

  v8f acc[2][4] = {};
  gemm_tile(arow0, arow1, b0, b1, b2, b3, IDIM, sub, acc);
#pragma unroll
  for (int mi = 0; mi < 2; ++mi)
#pragma unroll
    for (int v = 0; v < 8; ++v) {
      int slot = m0 + 16 * mi + v + 8 * sub;
      if (slot < count) {
        int p   = pair_list[e * CAP + slot];
        float w = w_list[e * CAP + slot];
        float* yrow = y_pair + (size_t)p * DIM;
#pragma unroll
        for (int ni = 0; ni < 4; ++ni)
          yrow[n0 + 16 * ni + lr] = w * acc[mi][ni][v];
      }
    }
}

// ---------------------------------------------------------------------------
// 7) Combine: out = sigmoid(x . egw) * shared + y_pair[t,0] + y_pair[t,1]
// ---------------------------------------------------------------------------
__global__ __launch_bounds__(256) void combine_kernel(
    const float* __restrict__ x, const float* __restrict__ egw,
    const float* __restrict__ y_pair, float* __restrict__ out) {
  __shared__ float red[256];
  __shared__ float sig;
  const int t = blockIdx.x;
  const int tid = threadIdx.x;

  float s = 0.f;
  for (int i = tid; i < DIM; i += 256) s += x[(size_t)t * DIM + i] * egw[i];
  red[tid] = s;
  __syncthreads();
  for (int st = 128; st > 0; st >>= 1) {
    if (tid < st) red[tid] += red[tid + st];
    __syncthreads();
  }
  if (tid == 0) sig = 1.0f / (1.0f + expf(-red[0]));
  __syncthreads();
  const float sg = sig;

  const float* y0 = y_pair + (size_t)(t * TOPK + 0) * DIM;
  const float* y1 = y_pair + (size_t)(t * TOPK + 1) * DIM;
  for (int i = tid; i < DIM; i += 256) {
    size_t o = (size_t)t * DIM + i;
    out[o] = sg * out[o] + y0[i] + y1[i];
  }
}

// ---------------------------------------------------------------------------
extern "C" void kernel_launch(void* const* d_in, const int* in_sizes, int n_in,
                              void* d_out, int out_size, void* d_ws,
                              size_t ws_size, hipStream_t stream) {
  const float* x   = (const float*)d_in[0];  // [T, D]
  const float* rw  = (const float*)d_in[1];  // [E, D]
  const float* gup = (const float*)d_in[2];  // [E, 2I, D]
  const float* dwn = (const float*)d_in[3];  // [E, D, I]
  const float* shg = (const float*)d_in[4];  // [SI, D]
  const float* shu = (const float*)d_in[5];  // [SI, D]
  const float* shd = (const float*)d_in[6];  // [D, SI]
  const float* egw = (const float*)d_in[7];  // [1, D]

  float* out   = (float*)d_out;                    // [T, D]
  float* probs = out + (size_t)T_TOK * DIM;        // [T, E]

  char* ws = (char*)d_ws;
  size_t off = 0;
  auto carve = [&](size_t bytes) -> void* {
    void* p = ws + off;
    off = (off + bytes + 255) & ~(size_t)255;
    return p;
  };
  int*   top_i     = (int*)  carve((size_t)T_TOK * TOPK * sizeof(int));
  float* top_w     = (float*)carve((size_t)T_TOK * TOPK * sizeof(float));
  int*   counts    = (int*)  carve((size_t)NEXP * sizeof(int));
  int*   tok_list  = (int*)  carve((size_t)NEXP * CAP * sizeof(int));
  int*   pair_list = (int*)  carve((size_t)NEXP * CAP * sizeof(int));
  float* w_list    = (float*)carve((size_t)NEXP * CAP * sizeof(float));
  float* act       = (float*)carve((size_t)NEXP * CAP * IDIM * sizeof(float)); // 32MB
  float* s1        = act;   // [T, SI] aliases act; consumed before experts run
  float* y_pair    = (float*)carve((size_t)T_TOK * TOPK * DIM * sizeof(float)); // 32MB

  router_kernel<<<T_TOK, 64, 0, stream>>>(x, rw, probs, top_i, top_w);
  dispatch_kernel<<<1, 64, 0, stream>>>(top_i, top_w, counts, tok_list,
                                        pair_list, w_list);
  {
    int n = T_TOK * TOPK * DIM;
    zero_kernel<<<(n + 255) / 256, 256, 0, stream>>>(y_pair, n);
  }
  shared_mlp1_kernel<<<dim3(SIDIM / 32, T_TOK / 32), 32, 0, stream>>>(x, shg,
                                                                      shu, s1);
  shared_mlp2_kernel<<<dim3(DIM / 64, T_TOK / 32), 32, 0, stream>>>(s1, shd,
                                                                    out);
  expert_mlp1_kernel<<<dim3(IDIM / 32, CAP / 32, NEXP), 32, 0, stream>>>(
      x, gup, counts, tok_list, act);
  expert_mlp2_kernel<<<dim3(DIM / 64, CAP / 32, NEXP), 32, 0, stream>>>(
      act, dwn, counts, pair_list, w_list, y_pair);
  combine_kernel<<<T_TOK, 256, 0, stream>>>(x, egw, y_pair, out);
}